// ComaGNNCritic_85023172591774
// MI455X (gfx1250) — compile-verified
//
#include <hip/hip_runtime.h>
#include <hip/hip_bf16.h>

typedef __attribute__((ext_vector_type(2))) float v2f;
typedef __attribute__((ext_vector_type(8))) float v8f;

// ---------------------------------------------------------------------------
// Build concatenated edge-conv weights (row-major, true sizes):
//   Wcat[k][n]      = W1[k][n] - W1[k+din][n]   (n < dout)
//   Wcat[k][dout+n] = W1[k+din][n]
//   cbias[n] = b1[n] for n < dout else 0
// ---------------------------------------------------------------------------
__global__ void prep_weights(const float* __restrict__ w1a, const float* __restrict__ b1a,
                             const float* __restrict__ w1b, const float* __restrict__ b1b,
                             const float* __restrict__ w1c, const float* __restrict__ b1c,
                             float* __restrict__ wcat1, float* __restrict__ cb1,
                             float* __restrict__ wcat2, float* __restrict__ cb2,
                             float* __restrict__ wcat3, float* __restrict__ cb3) {
  int idx = blockIdx.x * blockDim.x + threadIdx.x;
  const int S1 = 110 * 142, S1B = 142, S2 = 32 * 64, S2B = 64, S3 = 32 * 64, S3B = 64;
  if (idx < S1) {
    int k = idx / 142, n = idx % 142;
    wcat1[idx] = (n < 71) ? (w1a[k * 71 + n] - w1a[(k + 110) * 71 + n])
                          : w1a[(k + 110) * 71 + (n - 71)];
    return;
  }
  idx -= S1;
  if (idx < S1B) { cb1[idx] = (idx < 71) ? b1a[idx] : 0.f; return; }
  idx -= S1B;
  if (idx < S2) {
    int k = idx / 64, n = idx % 64;
    wcat2[idx] = (n < 32) ? (w1b[k * 32 + n] - w1b[(k + 32) * 32 + n])
                          : w1b[(k + 32) * 32 + (n - 32)];
    return;
  }
  idx -= S2;
  if (idx < S2B) { cb2[idx] = (idx < 32) ? b1b[idx] : 0.f; return; }
  idx -= S2B;
  if (idx < S3) {
    int k = idx / 64, n = idx % 64;
    wcat3[idx] = (n < 32) ? (w1c[k * 32 + n] - w1c[(k + 32) * 32 + n])
                          : w1c[(k + 32) * 32 + (n - 32)];
    return;
  }
  idx -= S3;
  if (idx < S3B) { cb3[idx] = (idx < 32) ? b1c[idx] : 0.f; return; }
}

// ---------------------------------------------------------------------------
// Pack a row-major (K x N) weight into WMMA fragment order with zero padding:
//   out[((nt*ksteps + ks)*32 + lane)*2 + {0,1}]
//     lane: col = nt*16 + (lane&15), k0 = ks*4 + (lane>>4)*2
// One thread per (nt, ks, lane); total threads = (Npad/16)*(Kpad/4)*32.
// ---------------------------------------------------------------------------
__global__ void pack_weights(const float* __restrict__ W, int K, int N,
                             int ksteps, int ntiles, float* __restrict__ out) {
  int idx = blockIdx.x * blockDim.x + threadIdx.x;
  int total = ntiles * ksteps * 32;
  if (idx >= total) return;
  int lane = idx & 31;
  int tmp = idx >> 5;
  int ks = tmp % ksteps;
  int nt = tmp / ksteps;
  int col = nt * 16 + (lane & 15);
  int k0 = ks * 4 + (lane >> 4) * 2;
  float v0 = (k0 < K && col < N) ? W[(size_t)k0 * N + col] : 0.f;
  float v1 = (k0 + 1 < K && col < N) ? W[(size_t)(k0 + 1) * N + col] : 0.f;
  out[(size_t)idx * 2]     = v0;
  out[(size_t)idx * 2 + 1] = v1;
}

// Zero-pad the input activations from (M x 110) to (M x 112).
__global__ void pad_input(const float* __restrict__ X0, float* __restrict__ Xp, int total) {
  int idx = blockIdx.x * blockDim.x + threadIdx.x;
  if (idx >= total) return;
  int r = idx / 112, c = idx - r * 112;
  Xp[idx] = (c < 110) ? X0[(size_t)r * 110 + c] : 0.f;
}

// ---------------------------------------------------------------------------
// C[M x N] = A[M x Kpad] @ Bpacked + bias (+relu). fp32 WMMA 16x16x4.
// One wave32 per (64-row x 16-col) strip: MREP=4 accumulators share each
// packed-B fragment. Inner loop: 1 b64 B-load + 4 b64 A-loads + 4 WMMAs,
// no conditionals (padding guarantees in-range zeros).
// ---------------------------------------------------------------------------
__global__ void gemm_wmma_f32(const float* __restrict__ A, int lda,
                              const float* __restrict__ Bp, int ksteps,
                              const float* __restrict__ bias, float* __restrict__ C,
                              int N, int do_relu) {
  const int lane = threadIdx.x;      // 0..31
  const int l16 = lane & 15;
  const int hi = lane >> 4;
  const int rowbase = blockIdx.x * 64;

  const float* aptr = A + (size_t)(rowbase + l16) * lda + hi * 2;
  const float* bptr = Bp + ((size_t)blockIdx.y * ksteps * 32 + lane) * 2;

  v8f acc0 = {0.f,0.f,0.f,0.f,0.f,0.f,0.f,0.f};
  v8f acc1 = acc0, acc2 = acc0, acc3 = acc0;

  for (int ks = 0; ks < ksteps; ++ks) {
    v2f b  = *(const v2f*)bptr;
    v2f a0 = *(const v2f*)(aptr);
    v2f a1 = *(const v2f*)(aptr + (size_t)16 * lda);
    v2f a2 = *(const v2f*)(aptr + (size_t)32 * lda);
    v2f a3 = *(const v2f*)(aptr + (size_t)48 * lda);
    acc0 = __builtin_amdgcn_wmma_f32_16x16x4_f32(false, a0, false, b, (short)0, acc0, false, false);
    acc1 = __builtin_amdgcn_wmma_f32_16x16x4_f32(false, a1, false, b, (short)0, acc1, false, false);
    acc2 = __builtin_amdgcn_wmma_f32_16x16x4_f32(false, a2, false, b, (short)0, acc2, false, false);
    acc3 = __builtin_amdgcn_wmma_f32_16x16x4_f32(false, a3, false, b, (short)0, acc3, false, false);
    aptr += 4;
    bptr += 64;
  }

  const int bcol = blockIdx.y * 16 + l16;
  if (bcol < N) {
    const float bv = bias ? bias[bcol] : 0.f;
    v8f accs[4] = {acc0, acc1, acc2, acc3};
#pragma unroll
    for (int r = 0; r < 4; ++r) {
#pragma unroll
      for (int v = 0; v < 8; ++v) {
        const int row = rowbase + r * 16 + (hi ? (v + 8) : v);
        float val = accs[r][v] + bv;
        if (do_relu) val = fmaxf(val, 0.f);
        C[(size_t)row * N + bcol] = val;
      }
    }
  }
}

// ---------------------------------------------------------------------------
// Per-graph edge kernel. P rows = (graph*8 + node), ld columns:
// cols [0,C) = A-part (bias folded), cols [C,2C) = B-part.
// edge (i,j): v = A_i + B_j ; LN(C) ; relu ; mean over j -> H (ldH cols,
// pad columns zeroed so H can feed a padded-K GEMM directly).
// ---------------------------------------------------------------------------
__global__ void edge_ln_mean(const float* __restrict__ P, int C, int ld,
                             const float* __restrict__ g, const float* __restrict__ be,
                             float* __restrict__ H, int ldH) {
  __shared__ float hbuf[64][72];
  const int bb = blockIdx.x;
  const int e = threadIdx.x;          // 0..63
  const int i = e >> 3, j = e & 7;
  const float* PA = P + (size_t)(bb * 8 + i) * ld;
  const float* PB = P + (size_t)(bb * 8 + j) * ld + C;

  float sum = 0.f, sq = 0.f;
  for (int c = 0; c < C; ++c) {
    float v = PA[c] + PB[c];
    hbuf[e][c] = v;
    sum += v;
    sq += v * v;
  }
  const float inv = 1.0f / (float)C;
  const float mean = sum * inv;
  const float var = sq * inv - mean * mean;
  const float rs = rsqrtf(var + 1e-5f);
  for (int c = 0; c < C; ++c) {
    float v = (hbuf[e][c] - mean) * rs * g[c] + be[c];
    hbuf[e][c] = fmaxf(v, 0.f);
  }
  __syncthreads();

  for (int idx = e; idx < 8 * ldH; idx += 64) {
    const int ni = idx / ldH, c = idx - ni * ldH;
    float out = 0.f;
    if (c < C) {
      float s = 0.f;
#pragma unroll
      for (int jj = 0; jj < 8; ++jj) s += hbuf[ni * 8 + jj][c];
      out = s * 0.125f;
    }
    H[(size_t)(bb * 8 + ni) * ldH + c] = out;
  }
}

// ---------------------------------------------------------------------------
// Head: p = mean over 8 nodes of Hh (16-dim); y = relu(p@phiW1+b1)@phiW2+b2.
// ---------------------------------------------------------------------------
__global__ void head_kernel(const float* __restrict__ Hh,
                            const float* __restrict__ phiW1, const float* __restrict__ phib1,
                            const float* __restrict__ phiW2, const float* __restrict__ phib2,
                            float* __restrict__ out) {
  __shared__ float p[16];
  __shared__ float q[16];
  const int bb = blockIdx.x, t = threadIdx.x;
  if (t < 16) {
    float s = 0.f;
    for (int nd = 0; nd < 8; ++nd) s += Hh[(size_t)(bb * 8 + nd) * 16 + t];
    p[t] = s * 0.125f;
  }
  __syncthreads();
  if (t < 16) {
    float s = phib1[t];
    for (int k = 0; k < 16; ++k) s += p[k] * phiW1[k * 16 + t];
    q[t] = fmaxf(s, 0.f);
  }
  __syncthreads();
  if (t == 0) {
    float s = phib2[0];
    for (int k = 0; k < 16; ++k) s += q[k] * phiW2[k];
    out[bb] = s;
  }
}

// ---------------------------------------------------------------------------
extern "C" void kernel_launch(void* const* d_in, const int* in_sizes, int n_in,
                              void* d_out, int out_size, void* d_ws, size_t ws_size,
                              hipStream_t stream) {
  (void)in_sizes; (void)n_in; (void)out_size; (void)ws_size;
  const float* X0     = (const float*)d_in[0];   // (32,128,8,110) -> (32768,110)
  const float* l1_W1  = (const float*)d_in[1];
  const float* l1_b1  = (const float*)d_in[2];
  const float* l1_g   = (const float*)d_in[3];
  const float* l1_be  = (const float*)d_in[4];
  const float* l1_W2  = (const float*)d_in[5];
  const float* l1_b2  = (const float*)d_in[6];
  const float* l2_W1  = (const float*)d_in[7];
  const float* l2_b1  = (const float*)d_in[8];
  const float* l2_g   = (const float*)d_in[9];
  const float* l2_be  = (const float*)d_in[10];
  const float* l2_W2  = (const float*)d_in[11];
  const float* l2_b2  = (const float*)d_in[12];
  const float* l3_W1  = (const float*)d_in[13];
  const float* l3_b1  = (const float*)d_in[14];
  const float* l3_g   = (const float*)d_in[15];
  const float* l3_be  = (const float*)d_in[16];
  const float* l3_W2  = (const float*)d_in[17];
  const float* l3_b2  = (const float*)d_in[18];
  const float* psi_W1 = (const float*)d_in[19];
  const float* psi_b1 = (const float*)d_in[20];
  const float* psi_W2 = (const float*)d_in[21];
  const float* psi_b2 = (const float*)d_in[22];
  const float* phi_W1 = (const float*)d_in[23];
  const float* phi_b1 = (const float*)d_in[24];
  const float* phi_W2 = (const float*)d_in[25];
  const float* phi_b2 = (const float*)d_in[26];

  const int M = 32768;               // 32*128*8 node rows

  // ---- workspace layout (floats) ----
  float* ws = (float*)d_ws;
  float* wcat1 = ws;                    // 110*142 = 15620
  float* cb1   = wcat1 + 15620;         // 144
  float* wcat2 = cb1 + 144;             // 2048
  float* cb2   = wcat2 + 2048;          // 64
  float* wcat3 = cb2 + 64;              // 2048
  float* cb3   = wcat3 + 2048;          // 64
  // packed weights (fragment order)
  float* p_w1   = cb3 + 64;             // 9*28*64 = 16128   (Kpad=112,Npad=144)
  float* p_l1W2 = p_w1 + 16128;         // 2*18*64 = 2304    (Kpad=72, Npad=32)
  float* p_w2   = p_l1W2 + 2304;        // 4*8*64  = 2048    (Kpad=32, Npad=64)
  float* p_l2W2 = p_w2 + 2048;          // 2*8*64  = 1024
  float* p_w3   = p_l2W2 + 1024;        // 2048
  float* p_l3W2 = p_w3 + 2048;          // 1024
  float* p_psi1 = p_l3W2 + 1024;        // 1*8*64  = 512     (Kpad=32, Npad=16)
  float* p_psi2 = p_psi1 + 512;         // 1*4*64  = 256     (Kpad=16, Npad=16)
  // big buffers (start at 49152 for alignment headroom)
  float* X0p = ws + 49152;              // 32768*112
  float* P   = X0p + (size_t)M * 112;   // 32768*142
  float* H   = P   + (size_t)M * 142;   // 32768*72
  float* X   = H   + (size_t)M * 72;    // 32768*32
  float* X5  = X   + (size_t)M * 32;    // 32768*16
  float* Hh  = X5  + (size_t)M * 16;    // 32768*16

  // ---- weight prep + packing (tiny) ----
  prep_weights<<<80, 256, 0, stream>>>(l1_W1, l1_b1, l2_W1, l2_b1, l3_W1, l3_b1,
                                       wcat1, cb1, wcat2, cb2, wcat3, cb3);
  pack_weights<<<32, 256, 0, stream>>>(wcat1, 110, 142, 28, 9, p_w1);
  pack_weights<<<5, 256, 0, stream>>>(l1_W2, 71, 32, 18, 2, p_l1W2);
  pack_weights<<<4, 256, 0, stream>>>(wcat2, 32, 64, 8, 4, p_w2);
  pack_weights<<<2, 256, 0, stream>>>(l2_W2, 32, 32, 8, 2, p_l2W2);
  pack_weights<<<4, 256, 0, stream>>>(wcat3, 32, 64, 8, 4, p_w3);
  pack_weights<<<2, 256, 0, stream>>>(l3_W2, 32, 32, 8, 2, p_l3W2);
  pack_weights<<<1, 256, 0, stream>>>(psi_W1, 32, 16, 8, 1, p_psi1);
  pack_weights<<<1, 256, 0, stream>>>(psi_W2, 16, 16, 4, 1, p_psi2);
  pad_input<<<(M * 112 + 255) / 256, 256, 0, stream>>>(X0, X0p, M * 112);

  const int GB = M / 64;   // 512 row strips

  // 1) layer-1 node projection: P = X0p @ wcat1 + cb1  (Kpad=112, N=142)
  gemm_wmma_f32<<<dim3(GB, 9), 32, 0, stream>>>(X0p, 112, p_w1, 28, cb1, P, 142, 0);
  edge_ln_mean<<<4096, 64, 0, stream>>>(P, 71, 142, l1_g, l1_be, H, 72);
  gemm_wmma_f32<<<dim3(GB, 2), 32, 0, stream>>>(H, 72, p_l1W2, 18, l1_b2, X, 32, 1);

  // 2) layer 2
  gemm_wmma_f32<<<dim3(GB, 4), 32, 0, stream>>>(X, 32, p_w2, 8, cb2, P, 64, 0);
  edge_ln_mean<<<4096, 64, 0, stream>>>(P, 32, 64, l2_g, l2_be, H, 32);
  gemm_wmma_f32<<<dim3(GB, 2), 32, 0, stream>>>(H, 32, p_l2W2, 8, l2_b2, X, 32, 1);

  // 3) layer 3 (no relu after)
  gemm_wmma_f32<<<dim3(GB, 4), 32, 0, stream>>>(X, 32, p_w3, 8, cb3, P, 64, 0);
  edge_ln_mean<<<4096, 64, 0, stream>>>(P, 32, 64, l3_g, l3_be, H, 32);
  gemm_wmma_f32<<<dim3(GB, 2), 32, 0, stream>>>(H, 32, p_l3W2, 8, l3_b2, X, 32, 0);

  // 4) psi MLP
  gemm_wmma_f32<<<dim3(GB, 1), 32, 0, stream>>>(X, 32, p_psi1, 8, psi_b1, X5, 16, 1);
  gemm_wmma_f32<<<dim3(GB, 1), 32, 0, stream>>>(X5, 16, p_psi2, 4, psi_b2, Hh, 16, 0);

  // 5) node mean + phi MLP -> (4096) outputs
  head_kernel<<<4096, 32, 0, stream>>>(Hh, phi_W1, phi_b1, phi_W2, phi_b2, (float*)d_out);
}